// RoIPooling_5669356833311
// MI455X (gfx1250) — compile-verified
//
#include <hip/hip_runtime.h>
#include <hip/hip_bf16.h>

// Problem constants (from reference)
#define IMG_SIZEF 224.0f
#define AFF       (2.0f / 7.0f)   // ROI / CROP
#define HH  56
#define WW  56
#define CC  256
#define LL  8
#define BB  64
#define KK  2048                  // L*C
#define NN  4096                  // OUT

typedef __attribute__((ext_vector_type(2))) float v2f;
typedef __attribute__((ext_vector_type(8))) float v8f;
typedef int v4i_gcc __attribute__((vector_size(4 * sizeof(int))));  // matches builtin proto

// ---------------------------------------------------------------------------
// Async global->LDS copy (16 bytes), tracked by ASYNCcnt.
// ---------------------------------------------------------------------------
__device__ __forceinline__ void async_cp16(const float* __restrict__ g,
                                           float* __restrict__ l) {
#if __has_builtin(__builtin_amdgcn_global_load_async_to_lds_b128)
  __builtin_amdgcn_global_load_async_to_lds_b128(
      (__attribute__((address_space(1))) v4i_gcc*)g,
      (__attribute__((address_space(3))) v4i_gcc*)l,
      0, 0);
#else
  unsigned loff = (unsigned)(size_t)(__attribute__((address_space(3))) void*)l;
  asm volatile("global_load_async_to_lds_b128 %0, %1, off"
               :: "v"(loff), "v"(g) : "memory");
#endif
}

__device__ __forceinline__ void wait_async0() {
#if __has_builtin(__builtin_amdgcn_s_wait_asynccnt)
  __builtin_amdgcn_s_wait_asynccnt(0);
#else
  asm volatile("s_wait_asynccnt 0" ::: "memory");
#endif
}

// ---------------------------------------------------------------------------
// Kernel 1: RoI pooling.  One block per (l, b), 256 threads = channels.
// Coordinates are block-uniform -> compiler scalarizes; each thread gathers
// 16 corner values (4 bilinear samples x 4 corners) and max-pools.
// ---------------------------------------------------------------------------
__global__ __launch_bounds__(CC)
void roi_pool_kernel(const float* __restrict__ feat,
                     const float* __restrict__ lms,
                     float* __restrict__ X) {
  const int l = blockIdx.x;
  const int b = blockIdx.y;
  const int c = threadIdx.x;

  const float lmx = lms[b * (2 * LL) + 2 * l];
  const float lmy = lms[b * (2 * LL) + 2 * l + 1];
  const float tx = -1.0f + 2.0f * lmx / IMG_SIZEF;
  const float ty = -1.0f + 2.0f * lmy / IMG_SIZEF;

  float ixs[2], iys[2];
#pragma unroll
  for (int i = 0; i < 2; ++i) {
    const float base = (i == 0) ? -0.5f : 0.5f;   // (2i+1)/ROI - 1
    const float gx = AFF * base + tx;
    const float gy = AFF * base + ty;
    ixs[i] = fminf(fmaxf(((gx + 1.0f) * (float)WW - 1.0f) * 0.5f, 0.0f), (float)(WW - 1));
    iys[i] = fminf(fmaxf(((gy + 1.0f) * (float)HH - 1.0f) * 0.5f, 0.0f), (float)(HH - 1));
  }

  const float* __restrict__ fbc = feat + (size_t)(b * CC + c) * (HH * WW);
  float best = -INFINITY;
#pragma unroll
  for (int sy = 0; sy < 2; ++sy) {
    const float iy = iys[sy];
    const int   y0 = (int)floorf(iy);
    const float wy = iy - (float)y0;
    const int   y1 = min(y0 + 1, HH - 1);
#pragma unroll
    for (int sx = 0; sx < 2; ++sx) {
      const float ix = ixs[sx];
      const int   x0 = (int)floorf(ix);
      const float wx = ix - (float)x0;
      const int   x1 = min(x0 + 1, WW - 1);
      const float v00 = fbc[y0 * WW + x0];
      const float v01 = fbc[y0 * WW + x1];
      const float v10 = fbc[y1 * WW + x0];
      const float v11 = fbc[y1 * WW + x1];
      const float val = (1.0f - wy) * ((1.0f - wx) * v00 + wx * v01) +
                        wy          * ((1.0f - wx) * v10 + wx * v11);
      best = fmaxf(best, val);
    }
  }
  X[(size_t)b * KK + l * CC + c] = best;
}

// ---------------------------------------------------------------------------
// Kernel 2: out = relu(X @ W^T + bias) with V_WMMA_F32_16X16X4_F32.
// Block = 128 threads (4 wave32s). blockIdx.y = M-tile (16 rows of X),
// blockIdx.x = 128-wide N block; each wave owns two 16-wide N tiles.
// K staged through double-buffered LDS in blocks of 32 via async
// global->LDS copies (ASYNCcnt), overlapping DMA with WMMA compute.
// Padded stride 36 floats (144 B): 16B-aligned b128 staging, and the b64
// fragment reads across 32 lanes cover all 64 banks conflict-free.
// ---------------------------------------------------------------------------
#define KB   32
#define LDSS (KB + 4)   // 36 floats = 144 B row stride
#define NBLK (KK / KB)  // 64 k-blocks

__global__ __launch_bounds__(128)
void gemm_relu_kernel(const float* __restrict__ X,     // 64 x 2048
                      const float* __restrict__ Wl,    // 4096 x 2048 (row-major)
                      const float* __restrict__ bias,  // 4096
                      float* __restrict__ out) {       // 64 x 4096
  __shared__ float As[2][16  * LDSS];
  __shared__ float Bs[2][128 * LDSS];

  const int tid  = threadIdx.x;
  const int wave = tid >> 5;
  const int lane = tid & 31;
  const int j    = lane & 15;       // N (for B/C/D) or M (for A) position
  const int hi   = lane >> 4;       // K-half selector within fragment

  const int M0 = blockIdx.y * 16;
  const int N0 = blockIdx.x * 128;
  const int nt0 = wave * 2;
  const int nt1 = wave * 2 + 1;

  // Staging coordinates (one b128 for A, eight b128 for B per thread)
  const int arow = tid >> 3;         // 0..15
  const int ac4  = (tid & 7) * 4;    // 0..28

  v8f acc0 = {};
  v8f acc1 = {};

  // Issue async stage for k-block starting at `kb` into buffer `buf`.
  auto stage = [&](int buf, int kb) {
    async_cp16(&X[(size_t)(M0 + arow) * KK + kb + ac4],
               &As[buf][arow * LDSS + ac4]);
#pragma unroll
    for (int i = 0; i < 8; ++i) {
      const int f   = tid + i * 128;   // 0..1023
      const int row = f >> 3;          // 0..127
      const int c4  = (f & 7) * 4;
      async_cp16(&Wl[(size_t)(N0 + row) * KK + kb + c4],
                 &Bs[buf][row * LDSS + c4]);
    }
  };

  stage(0, 0);   // prologue: fill buffer 0

  for (int it = 0; it < NBLK; ++it) {
    wait_async0();     // this wave's copies for buffer (it&1) have landed
    __syncthreads();   // everyone's copies landed; prev compute done

    if (it + 1 < NBLK) stage((it + 1) & 1, (it + 1) * KB);  // overlap with WMMA

    const float* __restrict__ as = &As[it & 1][0];
    const float* __restrict__ bs = &Bs[it & 1][0];
#pragma unroll
    for (int q = 0; q < KB / 4; ++q) {
      const int ko = 4 * q + 2 * hi;
      const v2f a  = *(const v2f*)&as[j * LDSS + ko];
      const v2f b0 = *(const v2f*)&bs[(nt0 * 16 + j) * LDSS + ko];
      const v2f b1 = *(const v2f*)&bs[(nt1 * 16 + j) * LDSS + ko];
      acc0 = __builtin_amdgcn_wmma_f32_16x16x4_f32(false, a, false, b0,
                                                   (short)0, acc0, false, false);
      acc1 = __builtin_amdgcn_wmma_f32_16x16x4_f32(false, a, false, b1,
                                                   (short)0, acc1, false, false);
    }
  }

  // Epilogue: bias (function of N only) + ReLU + store.
  const int n0 = N0 + nt0 * 16 + j;
  const int n1 = N0 + nt1 * 16 + j;
  const float bb0 = bias[n0];
  const float bb1 = bias[n1];
#pragma unroll
  for (int r = 0; r < 8; ++r) {
    const int m = M0 + r + 8 * hi;   // C/D layout: VGPR r -> M=r / M=r+8
    out[(size_t)m * NN + n0] = fmaxf(acc0[r] + bb0, 0.0f);
    out[(size_t)m * NN + n1] = fmaxf(acc1[r] + bb1, 0.0f);
  }
}

// ---------------------------------------------------------------------------
extern "C" void kernel_launch(void* const* d_in, const int* in_sizes, int n_in,
                              void* d_out, int out_size, void* d_ws, size_t ws_size,
                              hipStream_t stream) {
  const float* feat = (const float*)d_in[0];   // (64,256,56,56)
  const float* lms  = (const float*)d_in[1];   // (64,16)
  const float* Wl   = (const float*)d_in[2];   // (4096,2048)
  const float* bias = (const float*)d_in[3];   // (4096,)
  float* out = (float*)d_out;                  // (64,4096)
  float* X   = (float*)d_ws;                   // pooled activations 64x2048 (512 KB)

  roi_pool_kernel<<<dim3(LL, BB), CC, 0, stream>>>(feat, lms, X);
  gemm_relu_kernel<<<dim3(NN / 128, BB / 16), 128, 0, stream>>>(X, Wl, bias, out);
}